// GraphQNNHybrid_65481071399813
// MI455X (gfx1250) — compile-verified
//
#include <hip/hip_runtime.h>
#include <math.h>

typedef __attribute__((ext_vector_type(2))) float v2f;
typedef __attribute__((ext_vector_type(4))) float v4f;
typedef __attribute__((ext_vector_type(8))) float v8f;

// ---------------------------------------------------------------------------
// Kernel 1: stream-zero the whole output (134 MB). This is the bandwidth-
// dominant part (~5.8us at 23.3 TB/s). B128 nontemporal stores.
// ---------------------------------------------------------------------------
__global__ void qnn_zero_fill(float* __restrict__ out, long long n4, long long ntotal) {
  long long i = (long long)blockIdx.x * blockDim.x + threadIdx.x;
  const long long stride = (long long)gridDim.x * blockDim.x;
  v4f z = {0.f, 0.f, 0.f, 0.f};
  v4f* p = (v4f*)out;
  for (; i < n4; i += stride) {
    __builtin_nontemporal_store(z, p + i);
  }
  // defensive tail (out_size is divisible by 4 here, so normally a no-op)
  if (blockIdx.x == 0 && threadIdx.x == 0) {
    for (long long t = n4 * 4; t < ntotal; ++t) out[t] = 0.f;
  }
}

// ---------------------------------------------------------------------------
// Per-wave GEMV tile via full-precision V_WMMA_F32_16X16X4_F32.
//  A (16x4 f32): lane l in 0..15 -> row (rowBase+l), K = {k, k+1};
//                lane 16+l       -> row (rowBase+l), K = {k+2, k+3}.
//  B (4x16 f32): only column N=0 nonzero: lane 0 holds x[k],x[k+1];
//                lane 16 holds x[k+2],x[k+3].
//  D column 0:   lane 0 VGPR r = y[rowBase+r], lane 16 VGPR r = y[rowBase+8+r].
// Two interleaved accumulator chains hide the WMMA->WMMA RAW latency.
// ---------------------------------------------------------------------------
__device__ __forceinline__ void gemv_tanh_wmma(
    const float* __restrict__ W, const float* __restrict__ bias,
    const float* __restrict__ xin, float* __restrict__ yout,
    int M, int K, int waveId, int nWaves, int lane) {
  const int half = lane >> 4;   // 0: K pair {0,1}; 1: K pair {2,3}
  const int mrow = lane & 15;
  for (int rowBase = waveId * 16; rowBase < M; rowBase += nWaves * 16) {
    int rowA = rowBase + mrow;
    if (rowA >= M) rowA = M - 1;               // clamp (discarded at writeback)
    const float* Wrow = W + (long long)rowA * K;
    v8f acc0 = {0.f, 0.f, 0.f, 0.f, 0.f, 0.f, 0.f, 0.f};
    v8f acc1 = {0.f, 0.f, 0.f, 0.f, 0.f, 0.f, 0.f, 0.f};
    for (int k = 0; k < K; k += 8) {           // K is a multiple of 8 for all layers
      v2f a0, a1, b0, b1;
      a0.x = Wrow[k + 2 * half];
      a0.y = Wrow[k + 2 * half + 1];
      a1.x = Wrow[k + 4 + 2 * half];
      a1.y = Wrow[k + 4 + 2 * half + 1];
      b0.x = 0.f; b0.y = 0.f; b1.x = 0.f; b1.y = 0.f;
      if (mrow == 0) {                         // lanes 0 and 16 carry column N=0
        b0.x = xin[k + 2 * half];
        b0.y = xin[k + 2 * half + 1];
        b1.x = xin[k + 4 + 2 * half];
        b1.y = xin[k + 4 + 2 * half + 1];
      }
      acc0 = __builtin_amdgcn_wmma_f32_16x16x4_f32(false, a0, false, b0,
                                                   (short)0, acc0, false, false);
      acc1 = __builtin_amdgcn_wmma_f32_16x16x4_f32(false, a1, false, b1,
                                                   (short)0, acc1, false, false);
    }
    if (mrow == 0) {
#pragma unroll
      for (int r = 0; r < 8; ++r) {
        int row = rowBase + half * 8 + r;
        if (row < M) yout[row] = tanhf(acc0[r] + acc1[r] + bias[row]);
      }
    }
  }
}

// ---------------------------------------------------------------------------
// Kernel 2: full MLP chain (single workgroup, 32 waves) + quantum scatter.
// The 2^12 nonzero amplitudes: st[f] = r * (-i)^kmis, scattered at Gray(B),
// B = (f<<12) | fixed_bits, Gray(B) = B ^ (B>>1).
// ---------------------------------------------------------------------------
__global__ void __launch_bounds__(1024, 1)
qnn_hybrid_kernel(const float* __restrict__ x,
                  const float* __restrict__ W1, const float* __restrict__ b1,
                  const float* __restrict__ W2, const float* __restrict__ b2,
                  const float* __restrict__ W3, const float* __restrict__ b3,
                  const float* __restrict__ qp,
                  float* __restrict__ out) {
  __shared__ float xlds[1024];
  __shared__ float y1[512];
  __shared__ float y2[128];
  __shared__ float y3[24];
  __shared__ float cj[12];
  __shared__ float sj[12];
  __shared__ int   bitsh[24];

  const int tid = threadIdx.x;
  const int lane = tid & 31;
  const int waveId = tid >> 5;
  const int nWaves = 32;

  xlds[tid] = x[tid];
  __syncthreads();

  gemv_tanh_wmma(W1, b1, xlds, y1, 512, 1024, waveId, nWaves, lane);
  __syncthreads();
  gemv_tanh_wmma(W2, b2, y1, y2, 128, 512, waveId, nWaves, lane);
  __syncthreads();
  gemv_tanh_wmma(W3, b3, y2, y3, 24, 128, waveId, nWaves, lane);
  __syncthreads();

  if (tid < 24) {
    float v = y3[tid];
    out[tid] = v;                               // first tuple element: "current"
    bitsh[tid] = (v > 0.5f) ? 1 : 0;            // PauliX basis bits
  }
  if (tid < 12) {
    float th = qp[tid * 4] + qp[tid * 4 + 1] + qp[tid * 4 + 2] + qp[tid * 4 + 3];
    cj[tid] = cosf(0.5f * th);
    sj[tid] = sinf(0.5f * th);
  }
  __syncthreads();

  // hoist tables to registers
  int   bl[12];
  float cl[12], sl[12];
#pragma unroll
  for (int j = 0; j < 12; ++j) { bl[j] = bitsh[j]; cl[j] = cj[j]; sl[j] = sj[j]; }
  int bfixed = 0;
#pragma unroll
  for (int t = 0; t < 12; ++t) bfixed |= bitsh[12 + t] << (11 - t);

  float* __restrict__ qout = out + 24;          // complex64 region (re,im pairs)
#pragma unroll
  for (int u = 0; u < 4; ++u) {
    int f = tid * 4 + u;                        // 1024 threads x 4 = 2^12 states
    float r = 1.f;
    int kmis = 0;
#pragma unroll
    for (int j = 0; j < 12; ++j) {
      int fj = (f >> (11 - j)) & 1;
      if (fj == bl[j]) { r *= cl[j]; }
      else             { r *= sl[j]; kmis++; }  // factor = -i*sin
    }
    float re, im;
    switch (kmis & 3) {                         // multiply by (-i)^kmis
      case 0:  re = r;   im = 0.f; break;
      case 1:  re = 0.f; im = -r;  break;
      case 2:  re = -r;  im = 0.f; break;
      default: re = 0.f; im = r;   break;
    }
    int B = (f << 12) | bfixed;
    int idx = B ^ (B >> 1);                     // CNOT-ladder = Gray code
    v2f val; val.x = re; val.y = im;
    *((v2f*)(qout + 2 * (long long)idx)) = val; // 8B-aligned (base+96B)
  }
}

// ---------------------------------------------------------------------------
extern "C" void kernel_launch(void* const* d_in, const int* in_sizes, int n_in,
                              void* d_out, int out_size, void* d_ws, size_t ws_size,
                              hipStream_t stream) {
  const float* x  = (const float*)d_in[0];
  const float* W1 = (const float*)d_in[1];
  const float* b1 = (const float*)d_in[2];
  const float* W2 = (const float*)d_in[3];
  const float* b2 = (const float*)d_in[4];
  const float* W3 = (const float*)d_in[5];
  const float* b3 = (const float*)d_in[6];
  const float* qp = (const float*)d_in[7];
  float* out = (float*)d_out;

  const long long ntotal = (long long)out_size;   // 24 + 2*2^24 floats
  const long long n4 = ntotal / 4;

  // 1) zero the 134 MB output region (bandwidth-dominant step)
  qnn_zero_fill<<<2048, 256, 0, stream>>>(out, n4, ntotal);
  // 2) MLP (fp32 WMMA) + amplitude scatter; stream order => after zero-fill
  qnn_hybrid_kernel<<<1, 1024, 0, stream>>>(x, W1, b1, W2, b2, W3, b3, qp, out);
}